// VQPatternMatrixV7_80616536146005
// MI455X (gfx1250) — compile-verified
//
#include <hip/hip_runtime.h>
#include <hip/hip_bf16.h>
#include <math.h>

typedef __attribute__((ext_vector_type(2))) float v2f;
typedef __attribute__((ext_vector_type(8))) float v8f;

__device__ __forceinline__ v8f wmma_f32_16x16x4(v2f a, v2f b, v8f c) {
  // (neg_a, A, neg_b, B, c_mod, C, reuse_a, reuse_b)
  return __builtin_amdgcn_wmma_f32_16x16x4_f32(false, a, false, b, (short)0, c,
                                               false, false);
}

// ---------------------------------------------------------------------------
// Kernel 1: K_norm = l2norm(patterns)  [512 x 64]
// ---------------------------------------------------------------------------
__global__ __launch_bounds__(64) void vq_knorm_kernel(
    const float* __restrict__ patterns, float* __restrict__ knorm) {
  __shared__ float red[64];
  const int p = blockIdx.x, c = threadIdx.x;
  const float v = patterns[p * 64 + c];
  red[c] = v * v;
  __syncthreads();
  for (int s = 32; s > 0; s >>= 1) {
    if (c < s) red[c] += red[c + s];
    __syncthreads();
  }
  const float sc = 1.0f / fmaxf(sqrtf(red[0]), 1e-12f);
  knorm[p * 64 + c] = v * sc;
}

// ---------------------------------------------------------------------------
// Kernel 2: emb_table[p] = LayerNorm(patterns[p] @ w2 + b2)  [512 x 1024]
// (pattern_emb depends only on hard_idx -> precompute all 512 rows once)
// ---------------------------------------------------------------------------
__global__ __launch_bounds__(256) void vq_table_kernel(
    const float* __restrict__ patterns, const float* __restrict__ w2,
    const float* __restrict__ b2, const float* __restrict__ ln2g,
    const float* __restrict__ ln2b, float* __restrict__ table) {
  __shared__ float pr[64];
  __shared__ float rs[256];
  __shared__ float rq[256];
  const int p = blockIdx.x, t = threadIdx.x;
  if (t < 64) pr[t] = patterns[p * 64 + t];
  __syncthreads();
  float h[4];
  float s = 0.f, q = 0.f;
#pragma unroll
  for (int i = 0; i < 4; ++i) {
    const int j = t + i * 256;
    float a = b2[j];
    for (int c = 0; c < 64; ++c) a += pr[c] * w2[(size_t)c * 1024 + j];
    h[i] = a;
    s += a;
    q += a * a;
  }
  rs[t] = s;
  rq[t] = q;
  __syncthreads();
  for (int st = 128; st > 0; st >>= 1) {
    if (t < st) {
      rs[t] += rs[t + st];
      rq[t] += rq[t + st];
    }
    __syncthreads();
  }
  const float mu = rs[0] * (1.0f / 1024.0f);
  const float var = rq[0] * (1.0f / 1024.0f) - mu * mu;
  const float rstd = rsqrtf(var + 1e-5f);
#pragma unroll
  for (int i = 0; i < 4; ++i) {
    const int j = t + i * 256;
    table[(size_t)p * 1024 + j] = (h[i] - mu) * rstd * ln2g[j] + ln2b[j];
  }
}

// ---------------------------------------------------------------------------
// Kernel 3: fused GEMM1 + LN + L2 + logits-GEMM + argmax + one-hot + gather.
// 128 threads = 4 waves; each wave owns a 16-row tile (block = 64 rows).
// w1 chunk staged per block TRANSPOSED (+XOR swizzle) so every WMMA operand
// is a single contiguous ds_load_b64 into an aligned VGPR pair.
// ---------------------------------------------------------------------------
__global__ __launch_bounds__(128) void vq_main_kernel(
    const float* __restrict__ x, const float* __restrict__ w1,
    const float* __restrict__ b1, const float* __restrict__ ln1g,
    const float* __restrict__ ln1b, const float* __restrict__ knorm,
    const float* __restrict__ table, const int* __restrict__ ep,
    const int* __restrict__ te, float* __restrict__ o_emb,
    float* __restrict__ o_asn, float* __restrict__ o_log,
    float* __restrict__ o_idx, float* __restrict__ o_q) {
  // Block-shared: ws1t[64 cols][68] (transposed w1 chunk), kt[16][68], ln[2][64]
  // Per-wave: xs[16][68], qt[16][68], idxs[16], scs[16]
  __shared__ float ws1t[64 * 68];
  __shared__ float kt[16 * 68];
  __shared__ float lng[64];
  __shared__ float lnb[64];
  __shared__ float lds[4 * 2208];

  const int tid = threadIdx.x;
  const int wv = tid >> 5;
  const int lane = tid & 31;
  const int col = lane & 15;   // N index / A row index
  const int hl = lane >> 4;    // half-wave select
  const int kb = hl << 1;      // K sub-offset (16x16x4 f32 operand layout)

  float* xs = lds + wv * 2208;
  float* qt = xs + 1088;
  float* idxs = qt + 1088;
  float* scs = idxs + 16;

  const int m0 = blockIdx.x * 64 + wv * 16;

  // temperature (epoch/total are device scalars)
  const float epoch = (float)ep[0], total = (float)te[0];
  const float warm = total * 0.1f;
  float temp = 2.0f;
  if (!(epoch < warm)) {
    const float prog = (epoch - warm) / (total - warm);
    temp = fmaxf(0.7f, 2.0f - 1.3f * prog);
  }
  const float invT = 1.0f / temp;

  float b1v[4];
#pragma unroll
  for (int n = 0; n < 4; ++n) b1v[n] = b1[n * 16 + col];
  if (tid < 64) {
    lng[tid] = ln1g[tid];
    lnb[tid] = ln1b[tid];
  }

  const v8f vzero = {0.f, 0.f, 0.f, 0.f, 0.f, 0.f, 0.f, 0.f};
  v8f acc[4] = {vzero, vzero, vzero, vzero};

  const int abase = col * 68 + kb;  // A operand base (xs / qt)
  int bbase[4];                     // B operand bases in ws1t per n-tile
#pragma unroll
  for (int n = 0; n < 4; ++n) bbase[n] = (n * 16 + col) * 68 + kb;

  // -------- GEMM1: Qpre = x @ w1, 64-wide K chunks staged through LDS ------
  for (int kc = 0; kc < 1024; kc += 64) {
    __syncthreads();  // WAR vs previous chunk's readers (also covers ln stage)
    // per-wave: stage x tile rows
#pragma unroll 4
    for (int r = 0; r < 16; ++r) {
      const float2 v =
          *(const float2*)(x + (size_t)(m0 + r) * 1024 + kc + lane * 2);
      xs[r * 68 + lane * 2] = v.x;
      xs[r * 68 + lane * 2 + 1] = v.y;
    }
    // block-cooperative: stage w1 chunk transposed -> ws1t[c][k^swz]
    // element (c,k) at ws1t[c*68 + (k ^ (((c>>4)&3)<<2))]  (bank-conflict-free)
#pragma unroll
    for (int i = 0; i < 8; ++i) {
      const int f = tid + i * 128;   // float4 index, 1024 total
      const int kr = f >> 4;         // k row 0..63
      const int c4 = (f & 15) << 2;  // col group 0,4,..,60
      const int krs = kr ^ (((c4 >> 4) & 3) << 2);
      const float4 v = *(const float4*)(w1 + (size_t)(kc + kr) * 64 + c4);
      ws1t[(c4 + 0) * 68 + krs] = v.x;
      ws1t[(c4 + 1) * 68 + krs] = v.y;
      ws1t[(c4 + 2) * 68 + krs] = v.z;
      ws1t[(c4 + 3) * 68 + krs] = v.w;
    }
    // prefetch next x chunk (gfx1250 global_prefetch path)
    if (kc + 64 < 1024) {
      __builtin_prefetch(
          x + (size_t)(m0 + (lane & 15)) * 1024 + kc + 64 + hl * 32, 0, 0);
    }
    __syncthreads();
#pragma unroll 4
    for (int kk = 0; kk < 64; kk += 4) {
      const v2f a = *(const v2f*)(xs + abase + kk);
#pragma unroll
      for (int n = 0; n < 4; ++n) {
        const v2f b = *(const v2f*)(ws1t + bbase[n] + (kk ^ (n << 2)));
        acc[n] = wmma_f32_16x16x4(a, b, acc[n]);
      }
    }
  }
  __syncthreads();
  // D layout: VGPR r -> row (r + 8*hl), N = col
#pragma unroll
  for (int n = 0; n < 4; ++n)
#pragma unroll
    for (int r = 0; r < 8; ++r) {
      const int row = r + 8 * hl;
      qt[row * 68 + n * 16 + col] = acc[n][r] + b1v[n];
    }
  __syncthreads();

  // -------- LayerNorm(C=64): stats one row per lane 0..15 ------------------
  if (lane < 16) {
    const int r = lane;
    float s = 0.f, q = 0.f;
    for (int c = 0; c < 64; ++c) {
      const float v = qt[r * 68 + c];
      s += v;
      q += v * v;
    }
    const float mu = s * (1.0f / 64.0f);
    const float var = q * (1.0f / 64.0f) - mu * mu;
    const float rstd = rsqrtf(var + 1e-5f);
    float n2 = 0.f;
    for (int c = 0; c < 64; ++c) {
      const float y = (qt[r * 68 + c] - mu) * rstd * lng[c] + lnb[c];
      qt[r * 68 + c] = y;
      n2 += y * y;
    }
    scs[r] = 1.0f / fmaxf(sqrtf(n2), 1e-12f);
  }
  __syncthreads();
  // cooperative: write Q output (pre-L2) and scale qt in place (float4)
  {
    const int row = lane >> 1;
    const int cs = (lane & 1) * 32;
    const float sc = scs[row];
#pragma unroll
    for (int j = 0; j < 8; ++j) {
      float4 v = *(float4*)(qt + row * 68 + cs + j * 4);
      *(float4*)(o_q + (size_t)(m0 + row) * 64 + cs + j * 4) = v;
      v.x *= sc; v.y *= sc; v.z *= sc; v.w *= sc;
      *(float4*)(qt + row * 68 + cs + j * 4) = v;
    }
  }

  // -------- logits = Qn @ Knormᵀ, knorm tiles staged per block; argmax -----
  float bestv[8];
  int besti[8];
#pragma unroll
  for (int r = 0; r < 8; ++r) {
    bestv[r] = -3.402823466e38f;
    besti[r] = 0;
  }

  for (int pt = 0; pt < 32; ++pt) {
    __syncthreads();  // WAR on kt; first iteration also orders qt scale above
    {
      const int row = tid >> 3;
      const int cs = (tid & 7) * 8;
      const float* src = knorm + (size_t)(pt * 16 + row) * 64 + cs;
      *(float4*)(kt + row * 68 + cs) = *(const float4*)(src);
      *(float4*)(kt + row * 68 + cs + 4) = *(const float4*)(src + 4);
    }
    __syncthreads();
    v8f la = vzero;
#pragma unroll 4
    for (int kk = 0; kk < 64; kk += 4) {
      const v2f a = *(const v2f*)(qt + abase + kk);
      const v2f b = *(const v2f*)(kt + abase + kk);
      la = wmma_f32_16x16x4(a, b, la);
    }
#pragma unroll
    for (int r = 0; r < 8; ++r) {
      const int row = r + 8 * hl;
      const float v = la[r] * invT;
      o_log[(size_t)(m0 + row) * 512 + pt * 16 + col] = v;
      // argmax across the 16-lane N group (first-max tie semantics)
      float mv = v;
      int mi = pt * 16 + col;
#pragma unroll
      for (int m = 1; m < 16; m <<= 1) {
        const float ov = __shfl_xor(mv, m, 32);
        const int oi = __shfl_xor(mi, m, 32);
        if (ov > mv || (ov == mv && oi < mi)) {
          mv = ov;
          mi = oi;
        }
      }
      if (mv > bestv[r]) {  // strict >: earlier pattern wins ties
        bestv[r] = mv;
        besti[r] = mi;
      }
    }
  }

  if (col == 0) {
#pragma unroll
    for (int r = 0; r < 8; ++r) {
      const int row = r + 8 * hl;
      idxs[row] = (float)besti[r];
      o_idx[m0 + row] = (float)besti[r];
    }
  }
  __syncthreads();

  // -------- assignments: zero-fill then one-hot (per-wave store order) ------
  for (int r = 0; r < 16; ++r) {
    float4* dst = (float4*)(o_asn + (size_t)(m0 + r) * 512);
#pragma unroll
    for (int j = 0; j < 4; ++j)
      dst[lane + j * 32] = make_float4(0.f, 0.f, 0.f, 0.f);
  }
  if (col == 0) {
#pragma unroll
    for (int r = 0; r < 8; ++r) {
      const int row = r + 8 * hl;
      o_asn[(size_t)(m0 + row) * 512 + besti[r]] = 1.0f;
    }
  }

  // -------- pattern_emb: gather precomputed rows (table is L2-resident) -----
  for (int r = 0; r < 16; ++r) {
    const int idx = (int)idxs[r];
    const float4* src = (const float4*)(table + (size_t)idx * 1024);
    float4* dst = (float4*)(o_emb + (size_t)(m0 + r) * 1024);
#pragma unroll
    for (int j = 0; j < 8; ++j) dst[lane + j * 32] = src[lane + j * 32];
  }
}

// ---------------------------------------------------------------------------
extern "C" void kernel_launch(void* const* d_in, const int* in_sizes, int n_in,
                              void* d_out, int out_size, void* d_ws,
                              size_t ws_size, hipStream_t stream) {
  (void)in_sizes; (void)n_in; (void)out_size; (void)ws_size;
  const float* x = (const float*)d_in[0];
  const float* w1 = (const float*)d_in[1];
  const float* b1 = (const float*)d_in[2];
  const float* ln1g = (const float*)d_in[3];
  const float* ln1b = (const float*)d_in[4];
  const float* patterns = (const float*)d_in[5];
  const float* w2 = (const float*)d_in[6];
  const float* b2 = (const float*)d_in[7];
  const float* ln2g = (const float*)d_in[8];
  const float* ln2b = (const float*)d_in[9];
  const int* epoch = (const int*)d_in[10];
  const int* total = (const int*)d_in[11];

  float* ws = (float*)d_ws;
  float* knorm = ws;              // 512*64
  float* table = ws + 512 * 64;   // 512*1024

  float* out = (float*)d_out;
  float* o_emb = out;                               // 32768*1024
  float* o_asn = o_emb + (size_t)32768 * 1024;      // 32768*512
  float* o_log = o_asn + (size_t)32768 * 512;       // 32768*512
  float* o_idx = o_log + (size_t)32768 * 512;       // 32768
  float* o_q = o_idx + 32768;                       // 32768*64

  vq_knorm_kernel<<<512, 64, 0, stream>>>(patterns, knorm);
  vq_table_kernel<<<512, 256, 0, stream>>>(patterns, w2, b2, ln2g, ln2b, table);
  vq_main_kernel<<<512, 128, 0, stream>>>(x, w1, b1, ln1g, ln1b, knorm, table,
                                          epoch, total, o_emb, o_asn, o_log,
                                          o_idx, o_q);
}